// PolylineSubgraphNetwork_2104533975150
// MI455X (gfx1250) — compile-verified
//
#include <hip/hip_runtime.h>
#include <stddef.h>

// ---------------------------------------------------------------------------
// PolylineSubgraphNetwork for MI455X (gfx1250), wave32 + WMMA f32 16x16x4.
// ---------------------------------------------------------------------------

typedef float v2f __attribute__((ext_vector_type(2)));
typedef float v8f __attribute__((ext_vector_type(8)));

#define HD      64           // hidden width H
#define LN_EPS  1e-5f
#define LWAVES  4            // waves per block in the WMMA layer kernel
#define AT_STR  132          // padded LDS row stride (floats) for A tiles

// ---------------------------------------------------------------------------
// Layer 0: h = relu(LN(x @ W0 + b0)); fused atomic seg-max into segA.
// One thread per row (K is tiny: D_IN=9). Memory-bound anyway.
// ---------------------------------------------------------------------------
__global__ __launch_bounds__(256)
void pn_layer0(const float* __restrict__ x,  const float* __restrict__ W0,
               const float* __restrict__ b0, const float* __restrict__ g0,
               const float* __restrict__ be0, const int* __restrict__ ids,
               float* __restrict__ hout, float* __restrict__ seg,
               int N, int DIN)
{
    __shared__ float wl[16 * HD];
    __shared__ float bl[HD], gl[HD], bel[HD];
    const int t = threadIdx.x;
    for (int i = t; i < DIN * HD; i += 256) wl[i] = W0[i];
    if (t < HD) { bl[t] = b0[t]; gl[t] = g0[t]; bel[t] = be0[t]; }
    __syncthreads();

    const int row = blockIdx.x * 256 + t;
    if (row >= N) return;

    float z[HD];
#pragma unroll
    for (int j = 0; j < HD; ++j) z[j] = bl[j];

    const float* xr = x + (size_t)row * DIN;
    for (int k = 0; k < DIN; ++k) {
        const float xk = xr[k];
#pragma unroll
        for (int j = 0; j < HD; ++j) z[j] = fmaf(xk, wl[k * HD + j], z[j]);
    }

    float s = 0.f;
#pragma unroll
    for (int j = 0; j < HD; ++j) s += z[j];
    const float mu = s * (1.f / HD);
    float q = 0.f;
#pragma unroll
    for (int j = 0; j < HD; ++j) { const float d = z[j] - mu; q += d * d; }
    const float rstd = rsqrtf(q * (1.f / HD) + LN_EPS);

#pragma unroll
    for (int j = 0; j < HD; ++j)
        z[j] = fmaxf((z[j] - mu) * rstd * gl[j] + bel[j], 0.f);

    const int sid = ids[row];
    float4* hv = (float4*)(hout + (size_t)row * HD);
    unsigned* sp = (unsigned*)(seg + (size_t)sid * HD);
#pragma unroll
    for (int j = 0; j < HD; j += 4) {
        float4 v; v.x = z[j]; v.y = z[j + 1]; v.z = z[j + 2]; v.w = z[j + 3];
        hv[j >> 2] = v;
        atomicMax(sp + j + 0, __float_as_uint(v.x));
        atomicMax(sp + j + 1, __float_as_uint(v.y));
        atomicMax(sp + j + 2, __float_as_uint(v.z));
        atomicMax(sp + j + 3, __float_as_uint(v.w));
    }
}

// ---------------------------------------------------------------------------
// Layers 1/2: h' = relu(LN(concat[h, segprev[ids]] @ W + b)) with fused
// atomic seg-max into segnext. One wave per 16-row tile, WMMA f32 16x16x4,
// K = 128 (cols 0..63 = h, 64..127 = gathered seg-max), N-tiles = 4.
// In-place safe: reads only its own 16 rows of h, writes the same rows.
// ---------------------------------------------------------------------------
template <int STOREH>
__global__ __launch_bounds__(32 * LWAVES)
void pn_layerC(const float* __restrict__ hin, const float* __restrict__ segprev,
               const float* __restrict__ W,   const float* __restrict__ bias,
               const float* __restrict__ g,   const float* __restrict__ be,
               const int* __restrict__ ids,
               float* __restrict__ hout, float* __restrict__ segnext, int N)
{
    __shared__ float at[LWAVES][16 * AT_STR];   // 33,792 B

    const int lane = threadIdx.x & 31;
    const int wv   = threadIdx.x >> 5;
    const int half = lane >> 4;        // 0: K lo / rows 0..7   1: K hi / rows 8..15
    const int idx  = lane & 15;
    const int r0   = (blockIdx.x * LWAVES + wv) * 16;
    float* A = at[wv];

    // --- stage polyline ids for this tile (lanes 0..15 hold rows 0..15) ----
    int myid = 0;
    if (lane < 16) {
        const int r = r0 + lane;
        if (r < N) myid = ids[r];
    }

    // --- stage A tile: 16 rows x 128 cols, coalesced float4 loads ----------
#pragma unroll
    for (int i = 0; i < 8; ++i) {                 // h part (cols 0..63)
        const int flat = i * 128 + lane * 4;
        const int row = flat >> 6, col = flat & 63;
        const int r = r0 + row;
        float4 v = make_float4(0.f, 0.f, 0.f, 0.f);
        if (r < N) v = *(const float4*)(hin + (size_t)r * HD + col);
        *(float4*)(A + row * AT_STR + col) = v;
    }
#pragma unroll
    for (int i = 0; i < 8; ++i) {                 // gathered agg part (64..127)
        const int flat = i * 128 + lane * 4;
        const int row = flat >> 6, col = flat & 63;
        const int sgr = __shfl(myid, row, 32);
        const int r = r0 + row;
        float4 v = make_float4(0.f, 0.f, 0.f, 0.f);
        if (r < N) v = *(const float4*)(segprev + (size_t)sgr * HD + col);
        *(float4*)(A + row * AT_STR + HD + col) = v;
    }
    // LDS ops from the same wave are in-order (DScnt); only this wave touches
    // its A-tile region, so no workgroup barrier is required.

    // --- accumulators: bias per output column --------------------------------
    v8f acc[4];
#pragma unroll
    for (int t = 0; t < 4; ++t) {
        const float bv = bias[t * 16 + idx];
#pragma unroll
        for (int j = 0; j < 8; ++j) acc[t][j] = bv;
    }

    // --- K loop: 32 steps of 16x16x4 fp32 WMMA, 4 N-tiles ------------------
    const float* Abase = A + idx * AT_STR + 2 * half;
#pragma unroll
    for (int kb = 0; kb < 32; ++kb) {
        const v2f a = *(const v2f*)(Abase + kb * 4);     // A[row=idx][k, k+1]
        const int krow = kb * 4 + 2 * half;
#pragma unroll
        for (int t = 0; t < 4; ++t) {
            v2f b;
            b.x = W[(size_t)krow * HD + t * 16 + idx];    // B[k  ][n]
            b.y = W[(size_t)(krow + 1) * HD + t * 16 + idx]; // B[k+1][n]
            acc[t] = __builtin_amdgcn_wmma_f32_16x16x4_f32(
                false, a, false, b, (short)0, acc[t], false, false);
        }
    }

    // --- LayerNorm: rows j+8*half live across lanes idx=0..15 of this half --
    float sum[8];
#pragma unroll
    for (int j = 0; j < 8; ++j)
        sum[j] = acc[0][j] + acc[1][j] + acc[2][j] + acc[3][j];
#pragma unroll
    for (int m = 1; m < 16; m <<= 1)
#pragma unroll
        for (int j = 0; j < 8; ++j) sum[j] += __shfl_xor(sum[j], m, 32);

    float mean[8];
#pragma unroll
    for (int j = 0; j < 8; ++j) mean[j] = sum[j] * (1.f / HD);

    float q[8];
#pragma unroll
    for (int j = 0; j < 8; ++j) {
        float acc2 = 0.f;
#pragma unroll
        for (int t = 0; t < 4; ++t) {
            const float d = acc[t][j] - mean[j];
            acc2 = fmaf(d, d, acc2);
        }
        q[j] = acc2;
    }
#pragma unroll
    for (int m = 1; m < 16; m <<= 1)
#pragma unroll
        for (int j = 0; j < 8; ++j) q[j] += __shfl_xor(q[j], m, 32);

    float rstd[8];
#pragma unroll
    for (int j = 0; j < 8; ++j) rstd[j] = rsqrtf(q[j] * (1.f / HD) + LN_EPS);

    int segj[8];
#pragma unroll
    for (int j = 0; j < 8; ++j) segj[j] = __shfl(myid, 8 * half + j, 32);

    float gl[4], bel[4];
#pragma unroll
    for (int t = 0; t < 4; ++t) { gl[t] = g[t * 16 + idx]; bel[t] = be[t * 16 + idx]; }

    // --- affine + ReLU + store + fused segment-max atomics ------------------
#pragma unroll
    for (int j = 0; j < 8; ++j) {
        const int r = r0 + 8 * half + j;
        if (r >= N) continue;
        unsigned* srow = (unsigned*)(segnext + (size_t)segj[j] * HD);
        float* hrow = hout + (size_t)r * HD;
#pragma unroll
        for (int t = 0; t < 4; ++t) {
            const int col = t * 16 + idx;
            const float val =
                fmaxf((acc[t][j] - mean[j]) * rstd[j] * gl[t] + bel[t], 0.f);
            if (STOREH) hrow[col] = val;
            atomicMax(srow + col, __float_as_uint(val));
        }
    }
}

// ---------------------------------------------------------------------------
// Second output: arange(P) as int32 in the tail of d_out.
// ---------------------------------------------------------------------------
__global__ __launch_bounds__(256)
void pn_iota(int* __restrict__ out, int P)
{
    const int p = blockIdx.x * 256 + threadIdx.x;
    if (p < P) out[p] = p;
}

// ---------------------------------------------------------------------------
extern "C" void kernel_launch(void* const* d_in, const int* in_sizes, int n_in,
                              void* d_out, int out_size, void* d_ws, size_t ws_size,
                              hipStream_t stream)
{
    (void)n_in; (void)ws_size;
    const float* x    = (const float*)d_in[0];
    const int*   ids  = (const int*)d_in[1];
    // d_in[2] is P on-device; recover P on host from out_size = P*(H+1).
    const float* W0 = (const float*)d_in[3];
    const float* b0 = (const float*)d_in[4];
    const float* g0 = (const float*)d_in[5];
    const float* be0= (const float*)d_in[6];
    const float* W1 = (const float*)d_in[7];
    const float* b1 = (const float*)d_in[8];
    const float* g1 = (const float*)d_in[9];
    const float* be1= (const float*)d_in[10];
    const float* W2 = (const float*)d_in[11];
    const float* b2 = (const float*)d_in[12];
    const float* g2 = (const float*)d_in[13];
    const float* be2= (const float*)d_in[14];

    const int DIN = in_sizes[3] / HD;          // W0 is [D_IN, H]
    const int N   = in_sizes[0] / DIN;
    const int P   = out_size / (HD + 1);

    float* h    = (float*)d_ws;                      // N*H  (updated in place)
    float* segA = h + (size_t)N * HD;                // P*H
    float* segB = segA + (size_t)P * HD;             // P*H
    float* outf = (float*)d_out;                     // P*H final seg-max

    // 0.0f bit pattern == 0x0 -> memset works; empty polylines stay 0.
    hipMemsetAsync(segA, 0, (size_t)P * HD * sizeof(float), stream);
    hipMemsetAsync(segB, 0, (size_t)P * HD * sizeof(float), stream);
    hipMemsetAsync(outf, 0, (size_t)P * HD * sizeof(float), stream);

    pn_layer0<<<(N + 255) / 256, 256, 0, stream>>>(
        x, W0, b0, g0, be0, ids, h, segA, N, DIN);

    const int rowsPerBlk = 16 * LWAVES;
    const int blocks = (N + rowsPerBlk - 1) / rowsPerBlk;

    pn_layerC<1><<<blocks, 32 * LWAVES, 0, stream>>>(
        h, segA, W1, b1, g1, be1, ids, h, segB, N);        // in-place h

    pn_layerC<0><<<blocks, 32 * LWAVES, 0, stream>>>(
        h, segB, W2, b2, g2, be2, ids, h, outf, N);        // no h store

    pn_iota<<<(P + 255) / 256, 256, 0, stream>>>(
        (int*)d_out + (size_t)P * HD, P);
}